// _SimpleGINLayer_74053826117703
// MI455X (gfx1250) — compile-verified
//
#include <hip/hip_runtime.h>
#include <math.h>

typedef __attribute__((ext_vector_type(2))) float v2f;
typedef __attribute__((ext_vector_type(8))) float v8f;

#define D 128
#define EF 32

__device__ __forceinline__ v8f wmma_f32_k4(v2f a, v2f b, v8f c) {
  // 8 args: (neg_a, A, neg_b, B, c_mod, C, reuse_a, reuse_b)
  return __builtin_amdgcn_wmma_f32_16x16x4_f32(false, a, false, b, (short)0, c,
                                               false, false);
}

__device__ __forceinline__ float gelu_exact(float v) {
  return 0.5f * v * (1.0f + erff(v * 0.70710678118654752f));
}

// Swizzled transposed-W LDS indices. WT[c][k] = W[k][c]; the XOR swizzle keeps
// (kb, kb+1) contiguous (8B-aligned; swizzle is a multiple of 4 and never
// touches bit1) while spreading a B-fragment read across all 64 LDS banks.
__device__ __forceinline__ int wt_idx_k128(int c, int k) {  // K = 128 (W1/W2)
  return c * 128 + (k ^ ((c & 15) << 2));
}
__device__ __forceinline__ int wt_idx_k32(int c, int k) {   // K = 32 (W_e)
  return c * 32 + (k ^ (((c >> 1) & 7) << 2));
}

// ---------------------------------------------------------------------------
// Kernel 1: x = h  (residual folded into the scatter accumulator)
// ---------------------------------------------------------------------------
__global__ void init_x_kernel(const float* __restrict__ h,
                              float* __restrict__ x, int n4) {
  int i = blockIdx.x * blockDim.x + threadIdx.x;
  if (i < n4) ((float4*)x)[i] = ((const float4*)h)[i];
}

// ---------------------------------------------------------------------------
// Kernel 2: per 16-edge tile, msg = edge_attr@W_e (+b_e +h[src]) via fp32
// WMMA, then global_atomic_add_f32 scatter into x[dst].
// One wave = 16 edges x 128 outputs; 8 waves/block; W_e staged in LDS.
// ---------------------------------------------------------------------------
__global__ void __launch_bounds__(256)
edge_scatter_kernel(const float* __restrict__ h,
                    const int* __restrict__ edge_index,
                    const float* __restrict__ edge_attr,
                    const float* __restrict__ W_e,
                    const float* __restrict__ b_e, float* __restrict__ x,
                    int E) {
  __shared__ float WTe[EF * D];  // 16 KB, swizzled transpose of W_e
  const int tid = threadIdx.x;
  for (int i = tid; i < EF * D; i += 256) {
    const int k = i >> 7;   // row of W_e (coalesced global read along c)
    const int c = i & 127;  // col
    WTe[wt_idx_k32(c, k)] = W_e[i];
  }
  __syncthreads();

  const int lane = tid & 31;
  const int wave = tid >> 5;
  const int l16 = lane & 15;
  const int lh = lane >> 4;  // 16-lane half
  const int e0 = (blockIdx.x * 8 + wave) * 16;
  if (e0 >= E) return;

  if (e0 + 16 <= E) {
    // lanes 0..15 hold src of edge e0+lane; lanes 16..31 hold dst of e0+lane-16
    const int ml = edge_index[2 * (e0 + l16) + lh];

    v8f acc[8];
#pragma unroll
    for (int n = 0; n < 8; ++n)
      acc[n] = (v8f){0.f, 0.f, 0.f, 0.f, 0.f, 0.f, 0.f, 0.f};

    const int arow = (e0 + l16) * EF;
#pragma unroll
    for (int k = 0; k < 8; ++k) {  // K = 32 in steps of 4
      const int kb = 4 * k + 2 * lh;
      const v2f a = *(const v2f*)(edge_attr + arow + kb);  // A: V0={K0|K2},V1={K1|K3}
      v2f bf[8];
#pragma unroll
      for (int n = 0; n < 8; ++n)  // batched LDS fragment loads
        bf[n] = *(const v2f*)(&WTe[wt_idx_k32(16 * n + l16, kb)]);
#pragma unroll
      for (int n = 0; n < 8; ++n)  // back-to-back WMMAs
        acc[n] = wmma_f32_k4(a, bf[n], acc[n]);
    }

    // Unsigned node offsets -> zero-extend, SADDR-base + u32-voffset forms
    unsigned s_off[8], d_off[8];
#pragma unroll
    for (int j = 0; j < 8; ++j) {
      const int row = j + 8 * lh;  // C/D: VGPR j -> row j (lanes<16) / j+8
      s_off[j] = (unsigned)__shfl(ml, row, 32) << 7;       // src * D
      d_off[j] = (unsigned)__shfl(ml, row + 16, 32) << 7;  // dst * D
    }

#pragma unroll
    for (int n = 0; n < 8; ++n) {
      const unsigned c = 16u * n + (unsigned)l16;
      const float bv = b_e[c];
      float hv[8];
#pragma unroll
      for (int j = 0; j < 8; ++j)  // batch the 8 gathers into one clause
        hv[j] = h[s_off[j] + c];
#pragma unroll
      for (int j = 0; j < 8; ++j)
        atomicAdd(&x[d_off[j] + c], acc[n][j] + bv + hv[j]);
    }
  } else {
    // scalar tail: one edge per lane
    const int e = e0 + lane;
    if (e >= E) return;
    const unsigned s = (unsigned)edge_index[2 * e + 0] << 7;
    const unsigned d = (unsigned)edge_index[2 * e + 1] << 7;
    float ea[EF];
    for (int k = 0; k < EF; ++k) ea[k] = edge_attr[e * EF + k];
    for (int c = 0; c < D; ++c) {
      float acc = b_e[c];
      for (int k = 0; k < EF; ++k) acc += ea[k] * W_e[k * D + c];
      atomicAdd(&x[d + c], acc + h[s + c]);
    }
  }
}

// ---------------------------------------------------------------------------
// Kernels 3/4: Y = [gelu](X @ W + bias), X: nrows x 128, W: 128 x 128.
// One wave = 16 rows x 128 cols, fp32 WMMA K=4, 32 k-steps; W staged in LDS.
// ---------------------------------------------------------------------------
template <bool GELU>
__global__ void __launch_bounds__(256)
gemm128_kernel(const float* __restrict__ X, const float* __restrict__ W,
               const float* __restrict__ bias, float* __restrict__ Y,
               int nrows) {
  __shared__ float WT[D * D];  // 64 KB, swizzled transpose of W
  const int tid = threadIdx.x;
  for (int i = tid; i < D * D; i += 256) {
    const int k = i >> 7;
    const int c = i & 127;
    WT[wt_idx_k128(c, k)] = W[i];
  }
  __syncthreads();

  const int lane = tid & 31;
  const int wave = tid >> 5;
  const int l16 = lane & 15;
  const int lh = lane >> 4;
  const int r0 = (blockIdx.x * 8 + wave) * 16;
  if (r0 >= nrows) return;

  if (r0 + 16 <= nrows) {
    v8f acc[8];
#pragma unroll
    for (int n = 0; n < 8; ++n)
      acc[n] = (v8f){0.f, 0.f, 0.f, 0.f, 0.f, 0.f, 0.f, 0.f};

    const unsigned arow = (unsigned)(r0 + l16) << 7;
#pragma unroll 2
    for (int k = 0; k < 32; ++k) {
      const int kb = 4 * k + 2 * lh;
      const v2f a = *(const v2f*)(X + arow + kb);
      v2f bf[8];
#pragma unroll
      for (int n = 0; n < 8; ++n)  // batched LDS fragment loads
        bf[n] = *(const v2f*)(&WT[wt_idx_k128(16 * n + l16, kb)]);
#pragma unroll
      for (int n = 0; n < 8; ++n)  // back-to-back WMMAs
        acc[n] = wmma_f32_k4(a, bf[n], acc[n]);
    }

    const unsigned yrow = (unsigned)(r0 + 8 * lh) << 7;  // +(j<<7) per j
#pragma unroll
    for (int n = 0; n < 8; ++n) {
      const unsigned c = 16u * n + (unsigned)l16;
      const float bv = bias[c];
#pragma unroll
      for (int j = 0; j < 8; ++j) {
        float v = acc[n][j] + bv;
        if (GELU) v = gelu_exact(v);
        Y[yrow + ((unsigned)j << 7) + c] = v;
      }
    }
  } else {
    // scalar tail: one row per lane
    const int r = r0 + lane;
    if (r >= nrows) return;
    const unsigned rb = (unsigned)r << 7;
    for (int c = 0; c < D; ++c) {
      float v = bias[c];
      for (int k = 0; k < D; ++k) v += X[rb + k] * W[k * D + c];
      if (GELU) v = gelu_exact(v);
      Y[rb + c] = v;
    }
  }
}

// ---------------------------------------------------------------------------
extern "C" void kernel_launch(void* const* d_in, const int* in_sizes, int n_in,
                              void* d_out, int out_size, void* d_ws,
                              size_t ws_size, hipStream_t stream) {
  const float* h          = (const float*)d_in[0];
  const int*   edge_index = (const int*)d_in[1];
  const float* edge_attr  = (const float*)d_in[2];
  const float* W_e        = (const float*)d_in[3];
  const float* b_e        = (const float*)d_in[4];
  const float* W1         = (const float*)d_in[5];
  const float* b1         = (const float*)d_in[6];
  const float* W2         = (const float*)d_in[7];
  const float* b2         = (const float*)d_in[8];

  const int N = in_sizes[0] / D;   // 100000
  const int E = in_sizes[1] / 2;   // 1600000

  float* x = (float*)d_ws;          // N*D floats: h + agg
  float* t = x + (size_t)N * D;     // N*D floats: gelu(x@W1+b1)
  float* out = (float*)d_out;

  const int n4 = (N * D) / 4;
  init_x_kernel<<<(n4 + 255) / 256, 256, 0, stream>>>(h, x, n4);

  const long long etiles = ((long long)E + 15) / 16;
  const int eblocks = (int)((etiles + 7) / 8);
  edge_scatter_kernel<<<eblocks, 256, 0, stream>>>(h, edge_index, edge_attr,
                                                   W_e, b_e, x, E);

  const long long rtiles = ((long long)N + 15) / 16;
  const int rblocks = (int)((rtiles + 7) / 8);
  gemm128_kernel<true><<<rblocks, 256, 0, stream>>>(x, W1, b1, t, N);
  gemm128_kernel<false><<<rblocks, 256, 0, stream>>>(t, W2, b2, out, N);
}